// TripletLoss_47751446397214
// MI455X (gfx1250) — compile-verified
//
#include <hip/hip_runtime.h>
#include <hip/hip_bf16.h>
#include <math.h>

#define N_PARTS 62
#define M_SAMP  512
#define D_FEAT  256
#define KPOS    8
#define TILE    16
#define MARGIN  0.2f
#define BPITCH  33          // B-chunk LDS row pitch (floats): conflict-free reads

typedef float v2f __attribute__((ext_vector_type(2)));
typedef float v8f __attribute__((ext_vector_type(8)));

// ---- CDNA5 async global->LDS path (guarded; falls back to sync copies) ----
#if defined(__AMDGCN__) && defined(__has_builtin)
#if __has_builtin(__builtin_amdgcn_global_load_async_to_lds_b128)
#define HAVE_ASYNC 1
typedef int v4i __attribute__((vector_size(16)));
typedef __attribute__((address_space(1))) v4i gv4i_t;   // global (AS1)
typedef __attribute__((address_space(3))) v4i lv4i_t;   // LDS    (AS3)
#if __has_builtin(__builtin_amdgcn_s_wait_asynccnt)
#define WAIT_ASYNC(n) do { __builtin_amdgcn_s_wait_asynccnt(n); \
                           asm volatile("" ::: "memory"); } while (0)
#else
#define WAIT_ASYNC(n) asm volatile("s_wait_asynccnt %0" :: "n"(n) : "memory")
#endif
#endif
#endif
#ifndef HAVE_ASYNC
#define HAVE_ASYNC 0
#define WAIT_ASYNC(n) do {} while (0)
#endif

// Copy 16 bytes global -> LDS (async DMA on CDNA5, sync fallback otherwise).
__device__ __forceinline__ void ld16_to_lds(void* ldst, const void* gsrc) {
#if HAVE_ASYNC
    __builtin_amdgcn_global_load_async_to_lds_b128(
        (gv4i_t*)gsrc, (lv4i_t*)ldst, 0, 0);
#else
    *(float4*)ldst = *(const float4*)gsrc;
#endif
}

// ---------------------------------------------------------------------------
// ws layout (floats):
//   [0 .. 61]    sum_dist   [62 .. 123] hard_sum
//   [124 ..185]  full_sum   [186 ..247] full_num (integer-valued fp32)
//   [256 .. 256+62*512) x2 (squared norms per row)
// ---------------------------------------------------------------------------

__global__ void tl_init_kernel(float* acc) {
    if (threadIdx.x < 256) acc[threadIdx.x] = 0.0f;
}

// One wave per row: 32 lanes * (2 x float4) = 256 floats.
__global__ void __launch_bounds__(256)
tl_x2_kernel(const float* __restrict__ feat, float* __restrict__ x2) {
    const int row  = blockIdx.x * 8 + (threadIdx.x >> 5);
    const int lane = threadIdx.x & 31;
    const float4* rp = (const float4*)(feat + (size_t)row * D_FEAT);
    float4 a = rp[lane];
    float4 b = rp[32 + lane];
    float s = a.x*a.x + a.y*a.y + a.z*a.z + a.w*a.w
            + b.x*b.x + b.y*b.y + b.z*b.z + b.w*b.w;
    #pragma unroll
    for (int off = 16; off; off >>= 1) s += __shfl_xor(s, off);
    if (lane == 0) x2[row] = s;
}

// Stage one 16-row x 32-k B chunk into LDS: 128 x b128, coalesced per row.
__device__ __forceinline__ void stage_chunk(float (*dst)[BPITCH],
                                            const float* fpart,
                                            int colBase, int kBase, int lane) {
    #pragma unroll
    for (int i = 0; i < 4; ++i) {
        int c  = lane + 32 * i;          // 0..127
        int r  = c >> 3;                 // row 0..15
        int cc = c & 7;                  // 16B chunk within 32-float row
        ld16_to_lds(&dst[r][cc * 4],
                    fpart + (size_t)(colBase + r) * D_FEAT + kBase + cc * 4);
    }
}

// Grid: (M_SAMP/TILE, N_PARTS). Block: 256 threads = 8 waves (wave32).
__global__ void __launch_bounds__(256)
tl_main_kernel(const float* __restrict__ feat, const int* __restrict__ label,
               const float* __restrict__ x2g, float* __restrict__ acc) {
    const int p       = blockIdx.y;
    const int rowBase = blockIdx.x * TILE;
    const int tid     = threadIdx.x;
    const int wave    = tid >> 5;
    const int lane    = tid & 31;
    const int h       = lane >> 4;       // half-wave selector (ISA f32 WMMA layout)
    const int l       = lane & 15;

    __shared__ float    Xrow[TILE][D_FEAT + 4];
    __shared__ float    Bbuf[8][2][TILE][BPITCH];   // per-wave double buffer
    __shared__ float    x2all[M_SAMP];
    __shared__ int      labAll[M_SAMP];
    __shared__ float    diag[TILE][TILE + 1];
    __shared__ float    hp[TILE][KPOS];
    __shared__ float    maxHp[TILE];
    __shared__ unsigned minHnU[TILE];

    const float* fpart = feat + (size_t)p * M_SAMP * D_FEAT;

    // --- Async-stage anchor tile, x2 table, labels into LDS ---
    #pragma unroll
    for (int i = 0; i < 4; ++i) {                   // 16 rows x 256 = 1024 chunks
        int c = tid + 256 * i;
        int r = c >> 6;                             // 64 chunks per row
        int d = (c & 63) * 4;
        ld16_to_lds(&Xrow[r][d], fpart + (size_t)(rowBase + r) * D_FEAT + d);
    }
    if (tid < 128) {
        ld16_to_lds(&x2all[tid * 4],  x2g   + p * M_SAMP + tid * 4);
        ld16_to_lds(&labAll[tid * 4], label + p * M_SAMP + tid * 4);
    }
    if (tid < TILE) { maxHp[tid] = -1e30f; minHnU[tid] = 0x7F800000u; }
    WAIT_ASYNC(0);
    __syncthreads();

    // ---- Phase 1: diagonal 16x16 tile (wave 0) -> positive distances ----
    if (wave == 0) {
        v8f c = {};
        for (int k = 0; k < D_FEAT; k += 4) {
            v2f a;
            a.x = Xrow[l][k + 2*h];
            a.y = Xrow[l][k + 2*h + 1];
            c = __builtin_amdgcn_wmma_f32_16x16x4_f32(
                    false, a, false, a, (short)0, c, false, false);
        }
        #pragma unroll
        for (int r = 0; r < 8; ++r) {
            int   i  = r + 8*h;
            float d2 = x2all[rowBase + i] + x2all[rowBase + l] - 2.0f * c[r];
            d2 = fmaxf(d2, 0.0f);
            diag[i][l] = (d2 > 0.0f) ? sqrtf(d2) : 0.0f;
        }
    }
    __syncthreads();
    if (tid < TILE) {
        int   i  = tid, cnt = 0;
        int   li = labAll[rowBase + i];
        float mx = -1e30f;
        for (int j = 0; j < TILE; ++j) {
            if (labAll[rowBase + j] == li) {
                float v = diag[i][j];
                if (cnt < KPOS) hp[i][cnt] = v;
                ++cnt;
                mx = fmaxf(mx, v);
            }
        }
        maxHp[i] = mx;
    }
    __syncthreads();

    // ---- Phase 2: sweep all 512 columns; 8 waves x 4 column-tiles ----
    // B-chunks DMA'd to LDS (double-buffered) and overlapped with WMMA.
    float distSum = 0.0f, negSum = 0.0f, negCnt = 0.0f;
    for (int t = 0; t < 4; ++t) {
        const int colBase = (wave * 4 + t) * TILE;
        int buf = 0;
        stage_chunk(Bbuf[wave][0], fpart, colBase, 0, lane);
        v8f c = {};
        for (int kc = 0; kc < D_FEAT; kc += 32) {
            if (kc + 32 < D_FEAT) {
                stage_chunk(Bbuf[wave][buf ^ 1], fpart, colBase, kc + 32, lane);
                WAIT_ASYNC(4);          // drain current buffer, next stays in flight
            } else {
                WAIT_ASYNC(0);
            }
            float (*Bc)[BPITCH] = Bbuf[wave][buf];
            #pragma unroll
            for (int kk = 0; kk < 32; kk += 4) {
                v2f a, b;
                a.x = Xrow[l][kc + kk + 2*h];
                a.y = Xrow[l][kc + kk + 2*h + 1];
                b.x = Bc[l][kk + 2*h];
                b.y = Bc[l][kk + 2*h + 1];
                c = __builtin_amdgcn_wmma_f32_16x16x4_f32(
                        false, a, false, b, (short)0, c, false, false);
            }
            buf ^= 1;
        }
        #pragma unroll
        for (int r = 0; r < 8; ++r) {
            int   i  = r + 8*h;
            int   gi = rowBase + i;
            int   gj = colBase + l;
            float d2 = x2all[gi] + x2all[gj] - 2.0f * c[r];
            d2 = fmaxf(d2, 0.0f);
            float dist = (d2 > 0.0f) ? sqrtf(d2) : 0.0f;
            distSum += dist;
            if (labAll[gi] != labAll[gj]) {              // negative pair
                atomicMin(&minHnU[i], __float_as_uint(dist));
                #pragma unroll
                for (int q = 0; q < KPOS; ++q) {
                    float tq = MARGIN + hp[i][q] - dist;
                    if (tq > 0.0f) { negSum += tq; negCnt += 1.0f; }
                }
            }
        }
    }

    #pragma unroll
    for (int off = 16; off; off >>= 1) {
        distSum += __shfl_xor(distSum, off);
        negSum  += __shfl_xor(negSum,  off);
        negCnt  += __shfl_xor(negCnt,  off);
    }
    if (lane == 0) {
        atomicAdd(&acc[p],               distSum);
        atomicAdd(&acc[2 * N_PARTS + p], negSum);
        atomicAdd(&acc[3 * N_PARTS + p], negCnt);
    }
    __syncthreads();
    if (tid < TILE) {
        float hn   = __uint_as_float(minHnU[tid]);
        float hard = fmaxf(MARGIN + maxHp[tid] - hn, 0.0f);
        atomicAdd(&acc[N_PARTS + p], hard);
    }
}

__global__ void tl_finalize_kernel(const float* __restrict__ acc,
                                   float* __restrict__ out) {
    int p = threadIdx.x;
    if (p >= N_PARTS) return;
    float sumDist = acc[p];
    float hardSum = acc[N_PARTS + p];
    float fullSum = acc[2 * N_PARTS + p];
    float fullNum = acc[3 * N_PARTS + p];
    out[p]               = (fullNum == 0.0f) ? 0.0f : fullSum / fmaxf(fullNum, 1.0f);
    out[N_PARTS + p]     = hardSum / (float)M_SAMP;
    out[2 * N_PARTS + p] = sumDist / (float)(M_SAMP * M_SAMP);
    out[3 * N_PARTS + p] = fullNum;
}

extern "C" void kernel_launch(void* const* d_in, const int* in_sizes, int n_in,
                              void* d_out, int out_size, void* d_ws, size_t ws_size,
                              hipStream_t stream) {
    (void)in_sizes; (void)n_in; (void)out_size; (void)ws_size;
    const float* feat  = (const float*)d_in[0];
    const int*   label = (const int*)d_in[1];

    float* acc = (float*)d_ws;          // 256 floats of accumulators
    float* x2  = acc + 256;             // 62*512 squared norms

    tl_init_kernel<<<dim3(1), dim3(256), 0, stream>>>(acc);
    tl_x2_kernel<<<dim3((N_PARTS * M_SAMP) / 8), dim3(256), 0, stream>>>(feat, x2);
    tl_main_kernel<<<dim3(M_SAMP / TILE, N_PARTS), dim3(256), 0, stream>>>(
        feat, label, x2, acc);
    tl_finalize_kernel<<<dim3(1), dim3(64), 0, stream>>>(acc, (float*)d_out);
}